// HawkesAttention4_88347477278766
// MI455X (gfx1250) — compile-verified
//
#include <hip/hip_runtime.h>
#include <math.h>

// Problem constants (match reference)
#define BB   2
#define LL   512
#define DMM  256
#define HHH  4
#define DKK  64
#define DVV  64
#define TTT  8
#define PWW  8

// GEMM dims (all projections + output proj share these)
#define GM   1024        // B*L rows
#define GN   256         // H*DK = H*DV = DM
#define GK   256

typedef float v2f __attribute__((ext_vector_type(2)));
typedef float v8f __attribute__((ext_vector_type(8)));

__device__ __forceinline__ float gelu_exact(float x) {
    return 0.5f * x * (1.0f + erff(x * 0.70710678118654752440f));
}

// phi MLP: Linear(1->8) GELU Linear(8->8) GELU Linear(8->1), params for (type tt, head h)
__device__ __forceinline__ float phi_eval(int tt, int h, float delta,
    const float* sW0, const float* sB0, const float* sW1,
    const float* sB1, const float* sWf, const float* sBf) {
    const int base = (tt * HHH + h) * PWW;
    const int wb   = base * PWW;
    float h0[PWW];
#pragma unroll
    for (int w = 0; w < PWW; ++w)
        h0[w] = gelu_exact(fmaf(delta, sW0[base + w], sB0[base + w]));
    float out = sBf[tt * HHH + h];
#pragma unroll
    for (int u = 0; u < PWW; ++u) {
        float acc = sB1[base + u];
#pragma unroll
        for (int w = 0; w < PWW; ++w)
            acc = fmaf(sW1[wb + u * PWW + w], h0[w], acc);
        out = fmaf(sWf[base + u], gelu_exact(acc), out);
    }
    return out;
}

// ---------------- LayerNorm (pre-LN on q) : one wave per row ----------------
__global__ __launch_bounds__(256) void ln_kernel(const float* __restrict__ q,
                                                 const float* __restrict__ g,
                                                 const float* __restrict__ beta,
                                                 float* __restrict__ qn) {
    int row  = blockIdx.x * (blockDim.x >> 5) + (threadIdx.x >> 5);
    int lane = threadIdx.x & 31;
    if (row >= BB * LL) return;
    const float* x = q + row * DMM;
    float v[DMM / 32];
    float s = 0.f;
#pragma unroll
    for (int i = 0; i < DMM / 32; ++i) { v[i] = x[lane + 32 * i]; s += v[i]; }
#pragma unroll
    for (int m = 16; m >= 1; m >>= 1) s += __shfl_xor(s, m, 32);
    float mu = s * (1.0f / DMM);
    float vs = 0.f;
#pragma unroll
    for (int i = 0; i < DMM / 32; ++i) { float d = v[i] - mu; vs += d * d; }
#pragma unroll
    for (int m = 16; m >= 1; m >>= 1) vs += __shfl_xor(vs, m, 32);
    float inv = rsqrtf(vs * (1.0f / DMM) + 1e-6f);
    float* y = qn + row * DMM;
#pragma unroll
    for (int i = 0; i < DMM / 32; ++i) {
        int idx = lane + 32 * i;
        y[idx] = (v[i] - mu) * inv * g[idx] + beta[idx];
    }
}

// ---- fp32 WMMA GEMM, compile-time strides: C[1024,256] = A @ W (+ Cinit) ----
// Each wave computes a 64x16 strip (4 M-subtiles) so each B fragment is
// reused by 4 v_wmma_f32_16x16x4_f32. All loads in the fully unrolled K loop
// are immediate-offset addressing from 2 base pointers (no per-iter addr math).
// A fragment: lane(l15,khalf) holds A[tm+l15][k+2*khalf .. +1] (contiguous float2)
// B fragment: lane(l15,khalf) holds W[k+2*khalf][tn+l15], W[k+2*khalf+1][tn+l15]
// C/D layout: VGPR r <-> row tm + r + 8*khalf, col tn + l15
__global__ __launch_bounds__(128) void gemm64x16_f32_kernel(
    const float* __restrict__ A, const float* __restrict__ W,
    const float* __restrict__ Cinit, float* __restrict__ C) {
    const int lane  = threadIdx.x & 31;
    const int wave  = threadIdx.x >> 5;
    const int l15   = lane & 15;
    const int khalf = lane >> 4;
    const int tile  = blockIdx.x * 4 + wave;       // 0..255
    const int tm    = (tile >> 4) << 6;            // 64-row strip
    const int tn    = (tile & 15) << 4;

    v8f acc[4] = {{}, {}, {}, {}};
    if (Cinit) {
#pragma unroll
        for (int mt = 0; mt < 4; ++mt)
#pragma unroll
            for (int r = 0; r < 8; ++r)
                acc[mt][r] = Cinit[(tm + mt * 16 + r + 8 * khalf) * GN + tn + l15];
    }
    const float* A0 = A + (tm + l15) * GK + 2 * khalf;
    const float* Wb = W + 2 * khalf * GN + tn + l15;
#pragma unroll
    for (int kb = 0; kb < GK; kb += 4) {
        v2f bfrag;
        bfrag.x = Wb[kb * GN];
        bfrag.y = Wb[kb * GN + GN];
#pragma unroll
        for (int mt = 0; mt < 4; ++mt) {
            v2f a = *(const v2f*)(A0 + mt * 16 * GK + kb);
            acc[mt] = __builtin_amdgcn_wmma_f32_16x16x4_f32(false, a, false, bfrag,
                                                            (short)0, acc[mt], false, false);
        }
    }
#pragma unroll
    for (int mt = 0; mt < 4; ++mt)
#pragma unroll
        for (int r = 0; r < 8; ++r)
            C[(tm + mt * 16 + r + 8 * khalf) * GN + tn + l15] = acc[mt][r];
}

// ---------------- Hawkes flash attention: one wave per (b,h,i-tile) --------
__global__ __launch_bounds__(128) void hawkes_attn_kernel(
    const float* __restrict__ Qh, const float* __restrict__ Kh,
    const float* __restrict__ Vh,
    const float* __restrict__ t_in, const int* __restrict__ ctype,
    const float* __restrict__ W0, const float* __restrict__ b0,
    const float* __restrict__ W1, const float* __restrict__ b1,
    const float* __restrict__ Wf, const float* __restrict__ bf,
    float* __restrict__ AO) {
    __shared__ float sW0[TTT * HHH * PWW];
    __shared__ float sB0[TTT * HHH * PWW];
    __shared__ float sW1[TTT * HHH * PWW * PWW];
    __shared__ float sB1[TTT * HHH * PWW];
    __shared__ float sWf[TTT * HHH * PWW];
    __shared__ float sBf[TTT * HHH];
    __shared__ float sP[4][16 * 18];   // per-wave P tile, padded stride 18 (even, conflict-free)

    for (int i = threadIdx.x; i < TTT * HHH * PWW; i += blockDim.x) {
        sW0[i] = W0[i]; sB0[i] = b0[i]; sB1[i] = b1[i]; sWf[i] = Wf[i];
    }
    for (int i = threadIdx.x; i < TTT * HHH * PWW * PWW; i += blockDim.x) sW1[i] = W1[i];
    for (int i = threadIdx.x; i < TTT * HHH; i += blockDim.x) sBf[i] = bf[i];
    __syncthreads();

    const int lane  = threadIdx.x & 31;
    const int wv    = threadIdx.x >> 5;
    const int l15   = lane & 15;
    const int khalf = lane >> 4;
    const int itiles = LL / 16;
    const int gw = blockIdx.x * 4 + wv;
    const int it = gw % itiles;
    const int bh = gw / itiles;
    const int h  = bh % HHH;
    const int b  = bh / HHH;

    const int HD = HHH * DKK;        // 256 (== H*DV too)
    const float inv_scale = 0.125f;  // 1/sqrt(DK)
    const float NEG_INF = -__builtin_inff();
    float* Pld = sP[wv];

    // ---- hoist the 16 Q fragments (invariant across j-tiles) ----
    const float* Qrow = Qh + (b * LL + it * 16 + l15) * HD + h * DKK + 2 * khalf;
    v2f qf[16];
#pragma unroll
    for (int ki = 0; ki < 16; ++ki) qf[ki] = *(const v2f*)(Qrow + ki * 4);

    // per-lane row metadata (rows r + 8*khalf of this i-tile)
    float tI[8]; int cI[8];
#pragma unroll
    for (int r = 0; r < 8; ++r) {
        int i = it * 16 + r + 8 * khalf;
        tI[r] = t_in[b * LL + i];
        cI[r] = ctype[b * LL + i];
    }
    float mrow[8], lrow[8];
#pragma unroll
    for (int r = 0; r < 8; ++r) { mrow[r] = NEG_INF; lrow[r] = 0.f; }
    v8f o[4] = {{}, {}, {}, {}};

    for (int jt = 0; jt <= it; ++jt) {
        const int jrow = b * LL + jt * 16;
        // ---- S = Q x K^T (16 fp32 WMMAs over DK=64, immediate-offset K loads) ----
        const float* Kptr = Kh + (jrow + l15) * HD + h * DKK + 2 * khalf;
        const float* Vptr = Vh + jrow * HD + h * DVV + 2 * khalf * HD + l15;
        if (jt < it) {  // prefetch next j-tile K/V rows (global_prefetch_b8)
            __builtin_prefetch((const void*)(Kptr + 16 * HD), 0, 1);
            __builtin_prefetch((const void*)(Vptr + 16 * HD), 0, 1);
        }
        v8f s = {};
#pragma unroll
        for (int ki = 0; ki < 16; ++ki) {
            v2f kfrag = *(const v2f*)(Kptr + ki * 4);
            s = __builtin_amdgcn_wmma_f32_16x16x4_f32(false, qf[ki], false, kfrag,
                                                      (short)0, s, false, false);
        }
        // ---- phi modulation + causal mask ----
        const int j = jt * 16 + l15;
        const float tj = t_in[b * LL + j];
        const int cjv = ctype[b * LL + j];
        const bool vj = cjv < TTT;
        const int cj = cjv < 0 ? 0 : (cjv > TTT - 1 ? TTT - 1 : cjv);
        float phk[8];
#pragma unroll
        for (int r = 0; r < 8; ++r) {
            int i = it * 16 + r + 8 * khalf;
            float delta = tI[r] - tj;
            bool vi = cI[r] < TTT;
            int ci = cI[r] < 0 ? 0 : (cI[r] > TTT - 1 ? TTT - 1 : cI[r]);
            float pq = phi_eval(ci, h, delta, sW0, sB0, sW1, sB1, sWf, sBf);
            float pk = phi_eval(cj, h, delta, sW0, sB0, sW1, sB1, sWf, sBf);
            if (!(vi && vj)) { pq = 0.f; pk = 0.f; }
            phk[r] = pk;
            float sv = s[r] * pq * pk * inv_scale;
            if (j > i) sv = NEG_INF;
            s[r] = sv;
        }
        // ---- online softmax (16-lane shuffle reductions per row) ----
#pragma unroll
        for (int r = 0; r < 8; ++r) {
            float rm = s[r];
            rm = fmaxf(rm, __shfl_xor(rm, 1, 32));
            rm = fmaxf(rm, __shfl_xor(rm, 2, 32));
            rm = fmaxf(rm, __shfl_xor(rm, 4, 32));
            rm = fmaxf(rm, __shfl_xor(rm, 8, 32));
            float mnew = fmaxf(mrow[r], rm);
            float corr = __expf(mrow[r] - mnew);
            float e = __expf(s[r] - mnew);
            float es = e;
            es += __shfl_xor(es, 1, 32);
            es += __shfl_xor(es, 2, 32);
            es += __shfl_xor(es, 4, 32);
            es += __shfl_xor(es, 8, 32);
            lrow[r] = lrow[r] * corr + es;
            mrow[r] = mnew;
            o[0][r] *= corr; o[1][r] *= corr; o[2][r] *= corr; o[3][r] *= corr;
            // P = exp(s-m) * phiK, staged in LDS in C-layout
            Pld[(r + 8 * khalf) * 18 + l15] = e * phk[r];
        }
        asm volatile("" ::: "memory");   // LDS stores precede fragment reloads; DS is in-order per wave
        // ---- O += P x V (A-fragments from LDS, immediate-offset V loads) ----
#pragma unroll
        for (int d = 0; d < 4; ++d) {
#pragma unroll
            for (int kb = 0; kb < 16; kb += 4) {
                v2f a = *(const v2f*)(&Pld[l15 * 18 + kb + 2 * khalf]);
                v2f bb;
                bb.x = Vptr[kb * HD + d * 16];
                bb.y = Vptr[kb * HD + HD + d * 16];
                o[d] = __builtin_amdgcn_wmma_f32_16x16x4_f32(false, a, false, bb,
                                                             (short)0, o[d], false, false);
            }
        }
        asm volatile("" ::: "memory");
    }
    // ---- finalize: divide by softmax denominator, write head output ----
    float* Obase = AO + (b * LL + it * 16) * HD + h * DVV;
#pragma unroll
    for (int r = 0; r < 8; ++r) {
        float invl = 1.0f / lrow[r];
#pragma unroll
        for (int d = 0; d < 4; ++d)
            Obase[(r + 8 * khalf) * HD + d * 16 + l15] = o[d][r] * invl;
    }
}

extern "C" void kernel_launch(void* const* d_in, const int* in_sizes, int n_in,
                              void* d_out, int out_size, void* d_ws, size_t ws_size,
                              hipStream_t stream) {
    const float* q    = (const float*)d_in[0];
    const float* k    = (const float*)d_in[1];
    const float* v    = (const float*)d_in[2];
    const float* t_in = (const float*)d_in[3];
    const int*   c    = (const int*)d_in[4];
    // d_in[5] = mask : causal triu, reproduced analytically -> unused
    const float* ln_g = (const float*)d_in[6];
    const float* ln_b = (const float*)d_in[7];
    const float* Wq   = (const float*)d_in[8];
    const float* Wk   = (const float*)d_in[9];
    const float* Wv   = (const float*)d_in[10];
    const float* Wo   = (const float*)d_in[11];
    const float* W0   = (const float*)d_in[12];
    const float* b0   = (const float*)d_in[13];
    const float* W1   = (const float*)d_in[14];
    const float* b1   = (const float*)d_in[15];
    const float* Wf   = (const float*)d_in[16];
    const float* bf   = (const float*)d_in[17];
    float* out = (float*)d_out;

    float* ws = (float*)d_ws;
    float* qn = ws;
    float* Qh = qn + GM * GN;
    float* Kh = Qh + GM * GN;
    float* Vh = Kh + GM * GN;
    float* AO = Vh + GM * GN;               // total 5 MB of d_ws

    // 1) pre-LayerNorm on q
    ln_kernel<<<dim3(GM / 8), dim3(256), 0, stream>>>(q, ln_g, ln_b, qn);

    // 2) projections: 256 waves (64x16 strips), 4 waves/block -> 64 blocks
    dim3 gblocks(64);
    gemm64x16_f32_kernel<<<gblocks, 128, 0, stream>>>(qn, Wq, nullptr, Qh);
    gemm64x16_f32_kernel<<<gblocks, 128, 0, stream>>>(k,  Wk, nullptr, Kh);
    gemm64x16_f32_kernel<<<gblocks, 128, 0, stream>>>(v,  Wv, nullptr, Vh);

    // 3) Hawkes flash attention: 256 waves, 4 per block
    hawkes_attn_kernel<<<dim3(BB * HHH * (LL / 16) / 4), 128, 0, stream>>>(
        Qh, Kh, Vh, t_in, c, W0, b0, W1, b1, Wf, bf, AO);

    // 4) output projection + residual (residual preloaded into C fragment)
    gemm64x16_f32_kernel<<<gblocks, 128, 0, stream>>>(AO, Wo, q, out);
}